// GCN_50019189129603
// MI455X (gfx1250) — compile-verified
//
#include <hip/hip_runtime.h>
#include <hip/hip_bf16.h>

typedef __attribute__((ext_vector_type(2))) float v2f;
typedef __attribute__((ext_vector_type(8))) float v8f;

__device__ __forceinline__ void atomAddF(float* p, float v) {
    // hardware global_atomic_add_f32 (no CAS loop)
    unsafeAtomicAdd(p, v);
}

// ---------------- degree / norm ----------------
__global__ void k_deg_init(float* __restrict__ deg, int N) {
    int i = blockIdx.x * blockDim.x + threadIdx.x;
    if (i < N) deg[i] = 1.0f;   // self loop
}

__global__ void k_deg_count(const int* __restrict__ ei, int E, float* __restrict__ deg) {
    int e = blockIdx.x * blockDim.x + threadIdx.x;
    if (e < E) {
        int d = __builtin_nontemporal_load(ei + (size_t)E + e);  // stream, don't pollute L2
        atomAddF(&deg[d], 1.0f);
    }
}

__global__ void k_deg_rsqrt(float* __restrict__ deg, int N) {
    int i = blockIdx.x * blockDim.x + threadIdx.x;
    if (i < N) deg[i] = rsqrtf(deg[i]);   // deg >= 1 always
}

// ---------------- dense GEMM h = x @ W1 via V_WMMA_F32_16X16X4_F32 ----------------
// One wave handles a 16-row tile of x, producing a 16x16 tile of h (H == 16).
__global__ void k_gemm_wmma(const float* __restrict__ x,
                            const float* __restrict__ W1,
                            float* __restrict__ h,
                            int N, int K) {
    const int lane = threadIdx.x & 31;
    const int wave = threadIdx.x >> 5;
    const int tile = blockIdx.x * (blockDim.x >> 5) + wave;
    const int ntiles = (N + 15) >> 4;
    if (tile >= ntiles) return;                 // wave-uniform: EXEC stays all-1s

    const int m  = lane & 15;                   // row within tile (A), col (B/C/D)
    const int kb = (lane >> 4) << 1;            // K sub-offset: lanes 16-31 hold K+2,K+3
    int row = tile * 16 + m;
    if (row >= N) row = N - 1;                  // clamp (no divergence before WMMA)
    const float* __restrict__ xrow = x + (size_t)row * K;

    v8f c = {};
    for (int k = 0; k < K; k += 4) {
        v2f a, b;
        // A 16x4: VGPR0 = K+kb, VGPR1 = K+kb+1 (contiguous pair, read-once -> NT)
        a.x = __builtin_nontemporal_load(xrow + k + kb);
        a.y = __builtin_nontemporal_load(xrow + k + kb + 1);
        // B 4x16: row K+kb / K+kb+1, column n = m  (W1 row-major [K,16], hot -> RT)
        b.x = W1[(k + kb) * 16 + m];
        b.y = W1[(k + kb + 1) * 16 + m];
        c = __builtin_amdgcn_wmma_f32_16x16x4_f32(false, a, false, b,
                                                  (short)0, c, false, false);
    }
    // C/D 16x16: VGPR r -> row (r + (lane>>4)*8), col = m
    const int rbase = (lane >> 4) << 3;
#pragma unroll
    for (int r = 0; r < 8; ++r) {
        const int orow = tile * 16 + rbase + r;
        if (orow < N) h[(size_t)orow * 16 + m] = c[r];
    }
}

// ---------------- self-loop seed: h_agg[i][:] = h[i][:] * dinv[i]^2 ----------------
__global__ void k_selfloop16(const float* __restrict__ h, const float* __restrict__ dinv,
                             float* __restrict__ hagg, int N) {
    int t = blockIdx.x * blockDim.x + threadIdx.x;
    if (t < N * 16) {
        int i = t >> 4;
        float w = dinv[i];
        hagg[t] = h[t] * w * w;
    }
}

// ---------------- layer-1 edge scatter: 4 threads/edge, float4 each ----------------
__global__ void k_edge_agg16(const int* __restrict__ ei, int E,
                             const float* __restrict__ dinv,
                             const float* __restrict__ h,
                             float* __restrict__ hagg) {
    unsigned t = blockIdx.x * blockDim.x + threadIdx.x;
    unsigned e = t >> 2;
    if (e >= (unsigned)E) return;
    const int q = t & 3;
    const int s = __builtin_nontemporal_load(ei + e);              // index stream: NT
    const int d = __builtin_nontemporal_load(ei + (size_t)E + e);  // index stream: NT
    const float norm = dinv[s] * dinv[d];
    const float4 v = ((const float4*)(h + (size_t)s * 16))[q];     // hot table: RT
    float* dst = hagg + (size_t)d * 16 + q * 4;
    atomAddF(dst + 0, v.x * norm);
    atomAddF(dst + 1, v.y * norm);
    atomAddF(dst + 2, v.z * norm);
    atomAddF(dst + 3, v.w * norm);
}

// ---------------- layer 2 per-node: bias + ReLU + [16x2] matvec + self-loop seed ----
__global__ void k_layer2_node(const float* __restrict__ hagg,
                              const float* __restrict__ b1,
                              const float* __restrict__ W2,
                              const float* __restrict__ dinv,
                              float* __restrict__ h2,
                              float* __restrict__ out2, int N) {
    int i = blockIdx.x * blockDim.x + threadIdx.x;
    if (i >= N) return;
    float a0 = 0.f, a1 = 0.f;
#pragma unroll
    for (int j = 0; j < 16; ++j) {
        float v = __builtin_nontemporal_load(hagg + (size_t)i * 16 + j) + b1[j];
        v = v > 0.f ? v : 0.f;
        a0 += v * W2[j * 2 + 0];
        a1 += v * W2[j * 2 + 1];
    }
    h2[(size_t)i * 2 + 0] = a0;
    h2[(size_t)i * 2 + 1] = a1;
    const float w = dinv[i] * dinv[i];
    out2[(size_t)i * 2 + 0] = a0 * w;
    out2[(size_t)i * 2 + 1] = a1 * w;
}

// ---------------- layer-2 edge scatter: 1 thread/edge, 2 atomics ----------------
__global__ void k_edge_agg2(const int* __restrict__ ei, int E,
                            const float* __restrict__ dinv,
                            const float* __restrict__ h2,
                            float* __restrict__ out2) {
    int e = blockIdx.x * blockDim.x + threadIdx.x;
    if (e >= E) return;
    const int s = __builtin_nontemporal_load(ei + e);
    const int d = __builtin_nontemporal_load(ei + (size_t)E + e);
    const float norm = dinv[s] * dinv[d];
    const float2 v = ((const float2*)(h2 + (size_t)s * 2))[0];
    atomAddF(&out2[(size_t)d * 2 + 0], v.x * norm);
    atomAddF(&out2[(size_t)d * 2 + 1], v.y * norm);
}

// ---------------- bias + 2-class log_softmax ----------------
__global__ void k_finalize(const float* __restrict__ out2,
                           const float* __restrict__ b2,
                           float* __restrict__ out, int N) {
    int i = blockIdx.x * blockDim.x + threadIdx.x;
    if (i >= N) return;
    const float v0 = out2[(size_t)i * 2 + 0] + b2[0];
    const float v1 = out2[(size_t)i * 2 + 1] + b2[1];
    const float mx = fmaxf(v0, v1);
    const float lse = mx + logf(expf(v0 - mx) + expf(v1 - mx));
    __builtin_nontemporal_store(v0 - lse, out + (size_t)i * 2 + 0);
    __builtin_nontemporal_store(v1 - lse, out + (size_t)i * 2 + 1);
}

extern "C" void kernel_launch(void* const* d_in, const int* in_sizes, int n_in,
                              void* d_out, int out_size, void* d_ws, size_t ws_size,
                              hipStream_t stream) {
    const float* x  = (const float*)d_in[0];
    const int*   ei = (const int*)d_in[1];
    const float* W1 = (const float*)d_in[2];
    const float* b1 = (const float*)d_in[3];
    const float* W2 = (const float*)d_in[4];
    const float* b2 = (const float*)d_in[5];
    float* out = (float*)d_out;

    const int H    = in_sizes[3];            // 16
    const int Fin  = in_sizes[2] / H;        // 128
    const int N    = in_sizes[0] / Fin;      // 100000
    const int E    = in_sizes[1] / 2;        // 3200000
    (void)n_in; (void)out_size; (void)ws_size;

    // workspace layout (floats): dinv[N] | h[16N] | hagg[16N] | h2[2N] | out2[2N]
    float* ws   = (float*)d_ws;
    float* dinv = ws;
    float* h    = ws + (size_t)N;
    float* hagg = ws + (size_t)17 * N;
    float* h2   = ws + (size_t)33 * N;
    float* out2 = ws + (size_t)35 * N;

    const int B = 256;
    // 1) degrees -> dinv
    k_deg_init <<<(N + B - 1) / B, B, 0, stream>>>(dinv, N);
    k_deg_count<<<(E + B - 1) / B, B, 0, stream>>>(ei, E, dinv);
    k_deg_rsqrt<<<(N + B - 1) / B, B, 0, stream>>>(dinv, N);

    // 2) h = x @ W1 via WMMA (4 waves/block, one 16-row tile per wave)
    {
        const int ntiles = (N + 15) / 16;
        const int wpb = 4;
        k_gemm_wmma<<<(ntiles + wpb - 1) / wpb, wpb * 32, 0, stream>>>(x, W1, h, N, Fin);
    }

    // 3) self-loop seed + 4) layer-1 edge scatter
    k_selfloop16<<<((size_t)N * 16 + B - 1) / B, B, 0, stream>>>(h, dinv, hagg, N);
    k_edge_agg16<<<((size_t)E * 4 + B - 1) / B, B, 0, stream>>>(ei, E, dinv, h, hagg);

    // 5) layer-2 per-node transform + self-loop seed
    k_layer2_node<<<(N + B - 1) / B, B, 0, stream>>>(hagg, b1, W2, dinv, h2, out2, N);

    // 6) layer-2 edge scatter
    k_edge_agg2<<<(E + B - 1) / B, B, 0, stream>>>(ei, E, dinv, h2, out2);

    // 7) bias + log_softmax
    k_finalize<<<(N + B - 1) / B, B, 0, stream>>>(out2, b2, out, N);
}